// ModelGlobal_86912958202186
// MI455X (gfx1250) — compile-verified
//
#include <hip/hip_runtime.h>
#include <hip/hip_bf16.h>
#include <math.h>

#define M_DIM 400
#define C_DIM 8
#define W_DIM 100
#define D_DIM 300
#define H_DIM 100
#define R_K   25
#define MC    (M_DIM * C_DIM)   // 3200

typedef float v2f __attribute__((ext_vector_type(2)));
typedef float v8f __attribute__((ext_vector_type(8)));

// ---------------------------------------------------------------------------
// Kernel A: per-mention entity-context attention -> scores[m,c]
// one block (256 threads) per mention
// ---------------------------------------------------------------------------
__global__ void __launch_bounds__(256)
entity_context_kernel(const float* __restrict__ ctxt_vec,   // [M,W,D]
                      const float* __restrict__ cand_vec,   // [M,C,D]
                      const float* __restrict__ A_lin,      // [D]
                      const float* __restrict__ B_lin,      // [D]
                      float* __restrict__ scores)           // [M*C]
{
    __shared__ float candA[C_DIM * D_DIM];   // cand * A
    __shared__ float candR[C_DIM * D_DIM];   // raw cand
    __shared__ float uc[W_DIM * C_DIM];      // u_vec per (w,c)
    __shared__ float u[W_DIM];
    __shared__ float beta[W_DIM];
    __shared__ float cfB[D_DIM];
    __shared__ float red[256];

    const int m = blockIdx.x;
    const int t = threadIdx.x;
    const float* ctxt = ctxt_vec + (size_t)m * W_DIM * D_DIM;
    const float* cand = cand_vec + (size_t)m * C_DIM * D_DIM;

    for (int i = t; i < C_DIM * D_DIM; i += 256) {
        float cv = cand[i];
        candR[i] = cv;
        candA[i] = cv * A_lin[i % D_DIM];
    }
    __syncthreads();

    // u_vec[w,c] = <candA[c], ctxt[w]>
    for (int task = t; task < W_DIM * C_DIM; task += 256) {
        int w = task >> 3, c = task & 7;
        const float* cw = ctxt + (size_t)w * D_DIM;
        const float* ca = candA + c * D_DIM;
        float acc = 0.f;
#pragma unroll 4
        for (int d = 0; d < D_DIM; ++d) acc = fmaf(ca[d], cw[d], acc);
        uc[task] = acc;
    }
    __syncthreads();

    if (t < W_DIM) {
        float mx = uc[t * 8];
#pragma unroll
        for (int c = 1; c < C_DIM; ++c) mx = fmaxf(mx, uc[t * 8 + c]);
        u[t] = mx;
    }
    __syncthreads();

    // sketch = R-th largest of u (value at sorted index R-1, ties included)
    if (t < W_DIM) {
        float v = u[t];
        int gt = 0, eq = 0;
        for (int j = 0; j < W_DIM; ++j) {
            float x = u[j];
            gt += (x > v);
            eq += (x == v);
        }
        red[t] = (gt <= (R_K - 1) && gt + eq >= R_K) ? v : -3.4e38f;
    } else red[t] = -3.4e38f;
    __syncthreads();
    for (int s = 128; s > 0; s >>= 1) {
        if (t < s) red[t] = fmaxf(red[t], red[t + s]);
        __syncthreads();
    }
    float sketch = red[0];
    __syncthreads();

    // thresholded softmax over W
    if (t < W_DIM) {
        float uv = u[t];
        float mv = (uv > sketch) ? uv : (sketch - 50.0f);
        u[t] = mv;
        red[t] = mv;
    } else red[t] = -3.4e38f;
    __syncthreads();
    for (int s = 128; s > 0; s >>= 1) {
        if (t < s) red[t] = fmaxf(red[t], red[t + s]);
        __syncthreads();
    }
    float rmax = red[0];
    __syncthreads();
    if (t < W_DIM) {
        float e = __expf(u[t] - rmax);
        beta[t] = e;
        red[t] = e;
    } else red[t] = 0.f;
    __syncthreads();
    for (int s = 128; s > 0; s >>= 1) {
        if (t < s) red[t] += red[t + s];
        __syncthreads();
    }
    float dsum = red[0];
    __syncthreads();
    if (t < W_DIM) beta[t] *= (1.0f / dsum);
    __syncthreads();

    // ctxt_full[d] * B[d]
    for (int d = t; d < D_DIM; d += 256) {
        float acc = 0.f;
#pragma unroll 4
        for (int w = 0; w < W_DIM; ++w) acc = fmaf(beta[w], ctxt[(size_t)w * D_DIM + d], acc);
        cfB[d] = acc * B_lin[d];
    }
    __syncthreads();

    // scores[m,c] = <candR[c], cfB>
    if (t < C_DIM) {
        const float* cr = candR + t * D_DIM;
        float acc = 0.f;
#pragma unroll 4
        for (int d = 0; d < D_DIM; ++d) acc = fmaf(cr[d], cfB[d], acc);
        scores[m * C_DIM + t] = acc;
    }
}

// ---------------------------------------------------------------------------
// scaled X = flat * C_linear
// ---------------------------------------------------------------------------
__global__ void scale_x_kernel(const float* __restrict__ cand_vec,
                               const float* __restrict__ C_lin,
                               float* __restrict__ X)
{
    int i = blockIdx.x * blockDim.x + threadIdx.x;
    if (i < MC * D_DIM) X[i] = cand_vec[i] * C_lin[i % 300];
}

__global__ void zero_kernel(float* __restrict__ p, int n)
{
    int i = blockIdx.x * blockDim.x + threadIdx.x;
    if (i < n) p[i] = 0.f;
}

// ---------------------------------------------------------------------------
// Pairwise GEMM with WMMA f32 16x16x4:
//   up[i,j] = sum_d X[i,d] * Y[j,d] + scores[j]        (i,j in [0,3200))
//
// Register-blocked 2x2: each wave owns a 32x32 macro-tile (4 independent
// 16x16 WMMA accumulator chains), sharing A fragments across the two column
// tiles and B fragments across the two row tiles -> 4 b64 loads feed 4
// WMMAs (1:1), and the 4 independent chains keep the XDL pipe busy.
// 3200/32 = 100 macro-tiles per dim -> 10000 wave tasks -> 1250 blocks x 8
// waves.  K = 300 = 75 steps of 4.
// A fragment (16x4, ISA layout): lanes 0-15 hold K=0,1; lanes 16-31 K=2,3 for
// rows 0-15 -> one float2 load per lane per step. B (4x16) mirrors it from Y.
// ---------------------------------------------------------------------------
__global__ void __launch_bounds__(256)
pairwise_wmma_kernel(const float* __restrict__ X,       // [3200,300] scaled
                     const float* __restrict__ Y,       // [3200,300] raw flat
                     const float* __restrict__ scores,  // [3200]
                     float* __restrict__ up)            // [3200,3200]
{
    const int lane = threadIdx.x & 31;
    const int wid  = threadIdx.x >> 5;
    const int task = blockIdx.x * 8 + wid;          // 0..9999
    const int rowBlk = task / 100;                  // 0..99 (32-row macro tile)
    const int colBlk = task - rowBlk * 100;         // 0..99 (32-col macro tile)
    const int r0 = rowBlk * 32;
    const int n0 = colBlk * 32;
    const int l15 = lane & 15;
    const int kh  = (lane >> 4) * 2;                // lanes 16-31 take K+2,K+3

    const float* a0Ptr = X + (size_t)(r0 + l15) * D_DIM + kh;
    const float* a1Ptr = a0Ptr + (size_t)16 * D_DIM;
    const float* b0Ptr = Y + (size_t)(n0 + l15) * D_DIM + kh;
    const float* b1Ptr = b0Ptr + (size_t)16 * D_DIM;

    v8f acc00 = {0.f, 0.f, 0.f, 0.f, 0.f, 0.f, 0.f, 0.f};
    v8f acc01 = acc00, acc10 = acc00, acc11 = acc00;

#pragma unroll 3
    for (int k = 0; k < D_DIM; k += 4) {
        v2f a0 = *(const v2f*)(a0Ptr + k);
        v2f a1 = *(const v2f*)(a1Ptr + k);
        v2f b0 = *(const v2f*)(b0Ptr + k);
        v2f b1 = *(const v2f*)(b1Ptr + k);
        acc00 = __builtin_amdgcn_wmma_f32_16x16x4_f32(false, a0, false, b0,
                                                      (short)0, acc00, false, false);
        acc01 = __builtin_amdgcn_wmma_f32_16x16x4_f32(false, a0, false, b1,
                                                      (short)0, acc01, false, false);
        acc10 = __builtin_amdgcn_wmma_f32_16x16x4_f32(false, a1, false, b0,
                                                      (short)0, acc10, false, false);
        acc11 = __builtin_amdgcn_wmma_f32_16x16x4_f32(false, a1, false, b1,
                                                      (short)0, acc11, false, false);
    }

    // C/D layout: VGPR v holds row (v + 8*(lane>=16)), col = lane&15
    const int col0 = n0 + l15;
    const int col1 = col0 + 16;
    const float sc0 = scores[col0];
    const float sc1 = scores[col1];
    const int rbase0 = r0 + ((lane >> 4) << 3);
    const int rbase1 = rbase0 + 16;

    float* o00 = up + (size_t)rbase0 * MC + col0;
    float* o01 = up + (size_t)rbase0 * MC + col1;
    float* o10 = up + (size_t)rbase1 * MC + col0;
    float* o11 = up + (size_t)rbase1 * MC + col1;
#pragma unroll
    for (int v = 0; v < 8; ++v) {
        o00[(size_t)v * MC] = acc00[v] + sc0;
        o01[(size_t)v * MC] = acc01[v] + sc1;
        o10[(size_t)v * MC] = acc10[v] + sc0;
        o11[(size_t)v * MC] = acc11[v] + sc1;
    }
}

// ---------------------------------------------------------------------------
// LBP helpers
// ---------------------------------------------------------------------------
__global__ void sum_s_kernel(const float* __restrict__ msg,  // [3200,400]
                             float* __restrict__ s)          // [3200]
{
    int i = blockIdx.x * blockDim.x + threadIdx.x;
    if (i >= MC) return;
    const float* row = msg + (size_t)i * M_DIM;
    float acc = 0.f;
#pragma unroll 4
    for (int z = 0; z < M_DIM; ++z) acc += row[z];
    s[i] = acc;
}

__global__ void __launch_bounds__(256)
lbp_step_kernel(const float* __restrict__ up,   // [3200,3200] = [a,b][m,c]
                const float* __restrict__ s,    // [3200] = [m,c]
                float* __restrict__ msg)        // [a,b][m] = [3200,400]
{
    int t = blockIdx.x * blockDim.x + threadIdx.x;
    if (t >= M_DIM * M_DIM) return;
    const int a = t / M_DIM;
    const int m = t - a * M_DIM;
    const float* srow = s + m * C_DIM;

    float v[C_DIM];
    float mx = -3.4e38f;
#pragma unroll
    for (int b = 0; b < C_DIM; ++b) {
        const float* urow = up + (size_t)(a * C_DIM + b) * MC + m * C_DIM;
        float best = urow[0] + srow[0];
#pragma unroll
        for (int c = 1; c < C_DIM; ++c) best = fmaxf(best, urow[c] + srow[c]);
        v[b] = best;
        mx = fmaxf(mx, best);
    }
    float sum = 0.f;
#pragma unroll
    for (int b = 0; b < C_DIM; ++b) sum += __expf(v[b] - mx);
    const float lse = mx + __logf(sum);
    const bool self = (a == m);
#pragma unroll
    for (int b = 0; b < C_DIM; ++b) {
        float sel = self ? 0.0f : (v[b] - lse);
        size_t mi = (size_t)(a * C_DIM + b) * M_DIM + m;
        float old = msg[mi];
        msg[mi] = __logf(0.5f * __expf(sel) + 0.5f * __expf(old));
    }
}

// ---------------------------------------------------------------------------
// Final: global message, log-softmax over C, 2->100->1 MLP
// one wave per mention
// ---------------------------------------------------------------------------
__global__ void __launch_bounds__(32)
final_kernel(const float* __restrict__ msg,     // [3200,400]
             const float* __restrict__ scores,  // [3200]
             const float* __restrict__ pem,     // [M,C]
             const float* __restrict__ W1,      // [H,2]
             const float* __restrict__ b1,      // [H]
             const float* __restrict__ W2,      // [H]
             const float* __restrict__ b2,      // [1]
             float* __restrict__ out)           // [M,C]
{
    __shared__ float gm[C_DIM];
    const int m = blockIdx.x;
    const int t = threadIdx.x;
    if (t < C_DIM) {
        const float* row = msg + (size_t)(m * C_DIM + t) * M_DIM;
        float acc = 0.f;
#pragma unroll 4
        for (int z = 0; z < M_DIM; ++z) acc += row[z];
        gm[t] = acc + scores[m * C_DIM + t];
    }
    __syncthreads();
    if (t < C_DIM) {
        float mx = gm[0];
#pragma unroll
        for (int c = 1; c < C_DIM; ++c) mx = fmaxf(mx, gm[c]);
        float sum = 0.f;
#pragma unroll
        for (int c = 0; c < C_DIM; ++c) sum += __expf(gm[c] - mx);
        const float fgs = gm[t] - (mx + __logf(sum));
        const float x1  = pem[m * C_DIM + t];
        float acc = b2[0];
#pragma unroll 4
        for (int h = 0; h < H_DIM; ++h) {
            float hv = fmaf(W1[h * 2], fgs, fmaf(W1[h * 2 + 1], x1, b1[h]));
            hv = fmaxf(hv, 0.0f);
            acc = fmaf(W2[h], hv, acc);
        }
        out[m * C_DIM + t] = acc;
    }
}

// ---------------------------------------------------------------------------
// launch
// ---------------------------------------------------------------------------
extern "C" void kernel_launch(void* const* d_in, const int* in_sizes, int n_in,
                              void* d_out, int out_size, void* d_ws, size_t ws_size,
                              hipStream_t stream)
{
    // inputs (setup_inputs order)
    const float* ctxt_vec = (const float*)d_in[1];   // [M,W,D]
    const float* cand_vec = (const float*)d_in[3];   // [M,C,D]
    const float* p_e_m    = (const float*)d_in[4];   // [M,C]
    const float* A_lin    = (const float*)d_in[5];
    const float* B_lin    = (const float*)d_in[6];
    const float* C_lin    = (const float*)d_in[7];
    const float* W1       = (const float*)d_in[8];
    const float* b1       = (const float*)d_in[9];
    const float* W2       = (const float*)d_in[10];
    const float* b2       = (const float*)d_in[11];
    float* out = (float*)d_out;

    // workspace layout (floats)
    float* ws      = (float*)d_ws;
    float* scores  = ws;                           // 3200
    float* Xscaled = scores + MC;                  // 960000
    float* up      = Xscaled + (size_t)MC * D_DIM; // 3200*3200
    float* msg     = up + (size_t)MC * MC;         // 3200*400
    float* s_buf   = msg + (size_t)MC * M_DIM;     // 3200

    // A: entity-context scores
    entity_context_kernel<<<M_DIM, 256, 0, stream>>>(ctxt_vec, cand_vec, A_lin, B_lin, scores);

    // X = flat * C_linear
    {
        int n = MC * D_DIM;
        scale_x_kernel<<<(n + 255) / 256, 256, 0, stream>>>(cand_vec, C_lin, Xscaled);
    }

    // msg = 0
    {
        int n = MC * M_DIM;
        zero_kernel<<<(n + 255) / 256, 256, 0, stream>>>(msg, n);
    }

    // up = X @ flat^T + scores  (WMMA f32 16x16x4, 2x2 register blocking)
    {
        const int macroPerDim = MC / 32;                 // 100
        const int nTasks = macroPerDim * macroPerDim;    // 10000 waves
        pairwise_wmma_kernel<<<nTasks / 8, 256, 0, stream>>>(
            Xscaled, cand_vec, scores, up);
    }

    // 5 LBP iterations
    for (int it = 0; it < 5; ++it) {
        sum_s_kernel<<<(MC + 255) / 256, 256, 0, stream>>>(msg, s_buf);
        lbp_step_kernel<<<(M_DIM * M_DIM) / 256, 256, 0, stream>>>(up, s_buf, msg);
    }

    // final scores -> MLP
    final_kernel<<<M_DIM, 32, 0, stream>>>(msg, scores, p_e_m, W1, b1, W2, b2, out);
}